// CCQC_classifier_9620726743440
// MI455X (gfx1250) — compile-verified
//
#include <hip/hip_runtime.h>
#include <hip/hip_bf16.h>

// ---------------------------------------------------------------------------
// CCQC classifier:  out[b,q] = sum_a sign(a,q) * |(U @ x_norm[b])_a|^2
// U = fixed 1024x1024 unitary built from weights (only layer d=0 runs).
// x is real => Re/Im parts are two real f32 GEMMs -> V_WMMA_F32_16X16X4_F32.
// N-tile = 32 batches/block: halves L2 traffic on U vs 16, 128 blocks total.
// x tile staged with GLOBAL_LOAD_ASYNC_TO_LDS_B128 (ASYNCcnt DMA path).
// ---------------------------------------------------------------------------

#define NQ   10
#define DIM  1024          // 1 << NQ
#define NTILE 32           // batch rows per block
#define XS_STRIDE 1028     // padded LDS row stride (floats) to dodge bank conflicts

typedef __attribute__((ext_vector_type(2))) float f32x2;
typedef __attribute__((ext_vector_type(8))) float f32x8;

// Device-pass-only probe for the gfx1250 async LDS-load builtin; host pass
// (__has_builtin == 0 there) and unknown-name cases fall back to manual copy.
#if defined(__AMDGCN__) && __has_builtin(__builtin_amdgcn_global_load_async_to_lds_b128)
#define HAVE_ASYNC_LDS 1
#endif

// builtin signature (from device-pass diagnostic): param0 = v4i AS1*, param1 = v4i AS3*
typedef int v4i __attribute__((vector_size(16)));
typedef __attribute__((address_space(1))) v4i* gv4_ptr;   // global int4*
typedef __attribute__((address_space(3))) v4i* lv4_ptr;   // LDS int4*

// ------------------------------ complex helpers ----------------------------
__device__ __forceinline__ float2 cmul(float2 a, float2 b) {
  return make_float2(a.x * b.x - a.y * b.y, a.x * b.y + a.y * b.x);
}
__device__ __forceinline__ float2 cadd(float2 a, float2 b) {
  return make_float2(a.x + b.x, a.y + b.y);
}

struct CMat { float2 m00, m01, m10, m11; };   // [[m00,m01],[m10,m11]]

__device__ __forceinline__ CMat matmul2(const CMat A, const CMat B) {
  CMat R;
  R.m00 = cadd(cmul(A.m00, B.m00), cmul(A.m01, B.m10));
  R.m01 = cadd(cmul(A.m00, B.m01), cmul(A.m01, B.m11));
  R.m10 = cadd(cmul(A.m10, B.m00), cmul(A.m11, B.m10));
  R.m11 = cadd(cmul(A.m10, B.m01), cmul(A.m11, B.m11));
  return R;
}
__device__ __forceinline__ CMat rxmat(float t) {
  float c = __cosf(0.5f * t), s = __sinf(0.5f * t);
  CMat M;
  M.m00 = make_float2(c, 0.f);  M.m01 = make_float2(0.f, -s);
  M.m10 = make_float2(0.f, -s); M.m11 = make_float2(c, 0.f);
  return M;
}
__device__ __forceinline__ CMat rzmat(float t) {
  float c = __cosf(0.5f * t), s = __sinf(0.5f * t);
  CMat M;
  M.m00 = make_float2(c, -s);    M.m01 = make_float2(0.f, 0.f);
  M.m10 = make_float2(0.f, 0.f); M.m11 = make_float2(c, s);
  return M;
}

// single-qubit gate on bit position p (pairs are disjoint per thread)
__device__ __forceinline__ void apply1q(float2* st, const CMat M, int p, int tid) {
  const int mask = (1 << p) - 1;
  #pragma unroll
  for (int t = tid; t < DIM / 2; t += 256) {
    int lo = ((t & ~mask) << 1) | (t & mask);
    int hi = lo | (1 << p);
    float2 s0 = st[lo], s1 = st[hi];
    st[lo] = cadd(cmul(M.m00, s0), cmul(M.m01, s1));
    st[hi] = cadd(cmul(M.m10, s0), cmul(M.m11, s1));
  }
  __syncthreads();
}

// ---------------------------------------------------------------------------
// Kernel 1: build U column-by-column. Block k evolves e_k through the circuit
// in LDS (8 KB) and scatters column k into row-major U_re/U_im.
// ---------------------------------------------------------------------------
__global__ __launch_bounds__(256) void ccqc_build_U(
    const float* __restrict__ w,      // weights, layer d=0: w[i*5 + j]
    float* __restrict__ Ure,          // [DIM][DIM]
    float* __restrict__ Uim) {        // [DIM][DIM]
  __shared__ float2 st[DIM];
  const int tid = threadIdx.x;
  const int col = blockIdx.x;

  for (int i = tid; i < DIM; i += 256)
    st[i] = make_float2(i == col ? 1.f : 0.f, 0.f);
  __syncthreads();

  // rotation layer: RX(w0) -> RZ(w1) -> RX(w2) per qubit, fused to one 2x2
  #pragma unroll
  for (int qi = 0; qi < NQ; ++qi) {
    CMat M = matmul2(rxmat(w[qi * 5 + 2]), matmul2(rzmat(w[qi * 5 + 1]), rxmat(w[qi * 5 + 0])));
    apply1q(st, M, (NQ - 1) - qi, tid);
  }

  // entangler: cphase(j, j+7 mod 10) then RX on target
  const int js[NQ + 1] = {0, 7, 4, 1, 8, 5, 2, 9, 6, 3, 0};
  #pragma unroll
  for (int i = 0; i < NQ; ++i) {
    const int pa = (NQ - 1) - js[i];
    const int pb = (NQ - 1) - js[i + 1];
    const float phi = w[i * 5 + 3];
    const float2 ph = make_float2(__cosf(phi), __sinf(phi));
    #pragma unroll
    for (int t = tid; t < DIM; t += 256)
      if (((t >> pa) & (t >> pb)) & 1) st[t] = cmul(st[t], ph);
    __syncthreads();
    apply1q(st, rxmat(w[i * 5 + 4]), pb, tid);
  }

  for (int i = tid; i < DIM; i += 256) {
    Ure[(size_t)i * DIM + col] = st[i].x;
    Uim[(size_t)i * DIM + col] = st[i].y;
  }
}

// ---------------------------------------------------------------------------
// Kernel 2: per block, a 32-batch tile.  S_re/S_im = U_{re,im} @ x_tile via
// V_WMMA_F32_16X16X4_F32 (two N-subtiles per A fragment), fused PauliZ-sign
// epilogue + 1/||x||^2 scaling.
//   A tile (16x4 f32):  lane = M (l&15), vgpr pair = K, lane-half => K+2
//   B tile (4x16 f32):  lane = N (l&15), vgpr pair = K, lane-half => K+2
//   D tile (16x16 f32): N = l&15, M = vgpr + 8*(l>>4)
// ---------------------------------------------------------------------------
__global__ __launch_bounds__(256) void ccqc_qgemm(
    const float* __restrict__ x,      // [B][DIM] raw (unnormalized)
    const float* __restrict__ Ure,
    const float* __restrict__ Uim,
    float* __restrict__ out) {        // [B][NQ]
  __shared__ float xs[NTILE * XS_STRIDE];   // x tile, padded rows (~131.6 KB)
  __shared__ float part[NTILE][12];         // signed prob sums (padded)
  __shared__ float norm2[NTILE];
  __shared__ float ps[256];

  const int tid = threadIdx.x;
  const int b0  = blockIdx.x * NTILE;

  if (tid < NTILE * 12) ((float*)part)[tid] = 0.f;

  // stage x tile into LDS: async DMA (ASYNCcnt) if available, else float4
  for (int i = tid; i < NTILE * (DIM / 4); i += 256) {
    int n  = i >> 8;                 // DIM/4 = 256 float4 per row
    int m4 = i & 255;
    const float* gsrc = x + (size_t)(b0 + n) * DIM + (size_t)m4 * 4;
    float*       ldst = &xs[n * XS_STRIDE + m4 * 4];
#if defined(HAVE_ASYNC_LDS)
    __builtin_amdgcn_global_load_async_to_lds_b128(
        (gv4_ptr)(uintptr_t)gsrc, (lv4_ptr)(uintptr_t)ldst, 0, 0);
#else
    float4 v = *(const float4*)gsrc;
    ldst[0] = v.x; ldst[1] = v.y; ldst[2] = v.z; ldst[3] = v.w;
#endif
  }
#if defined(HAVE_ASYNC_LDS)
  asm volatile("s_wait_asynccnt 0x0" ::: "memory");
#endif
  __syncthreads();

  // per-row squared norms: 8 threads per row, then 32 threads finalize
  {
    int n = tid >> 3, s = tid & 7;
    float a = 0.f;
    #pragma unroll 8
    for (int u = 0; u < DIM / 8; ++u) {
      float v = xs[n * XS_STRIDE + s + 8 * u];
      a += v * v;
    }
    ps[tid] = a;
  }
  __syncthreads();
  if (tid < NTILE) {
    float a = 0.f;
    #pragma unroll
    for (int s = 0; s < 8; ++s) a += ps[tid * 8 + s];
    norm2[tid] = a;
  }

  // ------------------------- WMMA main loop --------------------------------
  const int wave = tid >> 5;
  const int lane = tid & 31;
  const int lm   = lane & 15;        // M (for A) / N (for B,D)
  const int half = lane >> 4;        // selects K+2 within the fragment

  const float* bbase0 = &xs[lm * XS_STRIDE + 2 * half];              // n = 0..15
  const float* bbase1 = &xs[(16 + lm) * XS_STRIDE + 2 * half];       // n = 16..31

  #pragma unroll 1
  for (int mt = 0; mt < 8; ++mt) {
    const int m0 = wave * 128 + mt * 16;          // amplitude-row tile base
    const float* are = Ure + (size_t)(m0 + lm) * DIM + 2 * half;
    const float* aim = Uim + (size_t)(m0 + lm) * DIM + 2 * half;

    if (mt < 7) {                                 // prefetch next A-tile rows
      __builtin_prefetch(are + 16 * DIM, 0, 1);
      __builtin_prefetch(aim + 16 * DIM, 0, 1);
    }

    f32x8 cre0 = {}, cim0 = {}, cre1 = {}, cim1 = {};
    #pragma unroll 4
    for (int k = 0; k < DIM; k += 4) {
      f32x2 a_re = *(const f32x2*)(are + k);
      f32x2 a_im = *(const f32x2*)(aim + k);
      f32x2 bb0  = *(const f32x2*)(bbase0 + k);
      f32x2 bb1  = *(const f32x2*)(bbase1 + k);
      cre0 = __builtin_amdgcn_wmma_f32_16x16x4_f32(false, a_re, false, bb0,
                                                   (short)0, cre0, false, false);
      cim0 = __builtin_amdgcn_wmma_f32_16x16x4_f32(false, a_im, false, bb0,
                                                   (short)0, cim0, false, false);
      cre1 = __builtin_amdgcn_wmma_f32_16x16x4_f32(false, a_re, false, bb1,
                                                   (short)0, cre1, false, false);
      cim1 = __builtin_amdgcn_wmma_f32_16x16x4_f32(false, a_im, false, bb1,
                                                   (short)0, cim1, false, false);
    }

    // epilogue: prob = re^2 + im^2, fold PauliZ signs per qubit
    float sq0[NQ], sq1[NQ];
    #pragma unroll
    for (int q = 0; q < NQ; ++q) { sq0[q] = 0.f; sq1[q] = 0.f; }
    #pragma unroll
    for (int v = 0; v < 8; ++v) {
      const int a = m0 + v + 8 * half;            // amplitude index of D[v]
      const float p0 = cre0[v] * cre0[v] + cim0[v] * cim0[v];
      const float p1 = cre1[v] * cre1[v] + cim1[v] * cim1[v];
      #pragma unroll
      for (int q = 0; q < NQ; ++q) {
        const bool neg = (a >> ((NQ - 1) - q)) & 1;
        sq0[q] += neg ? -p0 : p0;
        sq1[q] += neg ? -p1 : p1;
      }
    }
    #pragma unroll
    for (int q = 0; q < NQ; ++q) {
      atomicAdd(&part[lm][q],      sq0[q]);       // ds_add_f32
      atomicAdd(&part[16 + lm][q], sq1[q]);
    }
  }
  __syncthreads();

  if (tid < NTILE * NQ) {
    int n = tid / NQ, q = tid % NQ;
    out[(size_t)(b0 + n) * NQ + q] = part[n][q] / norm2[n];
  }
}

// ---------------------------------------------------------------------------
extern "C" void kernel_launch(void* const* d_in, const int* in_sizes, int n_in,
                              void* d_out, int out_size, void* d_ws, size_t ws_size,
                              hipStream_t stream) {
  const float* x   = (const float*)d_in[0];       // (4096,1,32,32) fp32
  const float* wts = (const float*)d_in[1];       // (DEPTH,10,5) fp32; d=0 slice used
  // d_in[2]/d_in[3] (weights_1/weights_2) are dead in the reference.

  float* Ure = (float*)d_ws;                      // 4 MB
  float* Uim = Ure + (size_t)DIM * DIM;           // 4 MB

  const int B = in_sizes[0] / DIM;                // 4096

  ccqc_build_U<<<DIM, 256, 0, stream>>>(wts, Ure, Uim);
  ccqc_qgemm<<<B / NTILE, 256, 0, stream>>>(x, Ure, Uim, (float*)d_out);
}